// MatCapNet_37434934952559
// MI455X (gfx1250) — compile-verified
//
#include <hip/hip_runtime.h>
#include <hip/hip_bf16.h>
#include <math.h>

#define EPSV 1e-8f

typedef float v2f __attribute__((ext_vector_type(2)));
typedef float v8f __attribute__((ext_vector_type(8)));

// ---------------------------------------------------------------------------
// Kernel 1: conv1 (5x5, stride 2, VALID) + bias + relu
// x: (8,3,32,32)  w: (32,3,5,5)  -> h: (8,32,14,14)
// ---------------------------------------------------------------------------
__global__ void conv1_relu_kernel(const float* __restrict__ x,
                                  const float* __restrict__ w,
                                  const float* __restrict__ bias,
                                  float* __restrict__ h)
{
    int idx = blockIdx.x * blockDim.x + threadIdx.x;
    const int total = 8 * 32 * 14 * 14;
    if (idx >= total) return;
    int xo = idx % 14;
    int yo = (idx / 14) % 14;
    int o  = (idx / 196) % 32;
    int b  = idx / (196 * 32);
    float s = bias[o];
    #pragma unroll
    for (int c = 0; c < 3; ++c) {
        #pragma unroll
        for (int ky = 0; ky < 5; ++ky) {
            #pragma unroll
            for (int kx = 0; kx < 5; ++kx) {
                float xv = x[(((size_t)b * 3 + c) * 32 + (yo * 2 + ky)) * 32 + (xo * 2 + kx)];
                float wv = w[(((size_t)o * 3 + c) * 5 + ky) * 5 + kx];
                s = fmaf(xv, wv, s);
            }
        }
    }
    h[idx] = fmaxf(s, 0.0f);
}

// ---------------------------------------------------------------------------
// Kernel 2: primary caps via f32 WMMA (V_WMMA_F32_16X16X4_F32).
// pose[pos, o] = sum_c h[b,c,y,x] * pc_pose_w[o,c]  (pos = b*196+y*14+x)
// act [pos, o] = sigmoid(sum_c h * pc_act_w[o,c] + b)
// One wave (32 threads) computes one 16(pos) x 16(out) tile; K=32 as 8 WMMA
// steps of K=4.  grid.y 0..31 -> pose n-tiles (512 outs); 32..33 -> act head.
// Layouts per ISA 7.12.2:
//   A 16x4 f32 : lane holds row M=lane&15; vgpr v + 2*(lane>>4) = K
//   B  4x16 f32: lane holds col N=lane&15; vgpr v + 2*(lane>>4) = K
//   C/D 16x16  : vgpr r, lane l -> M = r + 8*(l>>4), N = l&15
// ---------------------------------------------------------------------------
__global__ void primary_caps_kernel(const float* __restrict__ h,
                                    const float* __restrict__ pose_w,
                                    const float* __restrict__ pose_b,
                                    const float* __restrict__ act_w,
                                    const float* __restrict__ act_b,
                                    float* __restrict__ M1,   // [pos][32][16]
                                    float* __restrict__ a1)   // [pos][32]
{
    const int lane  = threadIdx.x;            // one wave per block
    const int tile  = blockIdx.x;             // 98 tiles of 16 positions
    const bool isAct = (blockIdx.y >= 32);
    const int nbase = isAct ? ((int)blockIdx.y - 32) * 16 : (int)blockIdx.y * 16;
    const float* Wsrc = isAct ? act_w : pose_w;

    const int mrow = lane & 15;
    const int hi2  = lane >> 4;
    const int posA = tile * 16 + mrow;        // A-row = flat position
    const int bA   = posA / 196;
    const int rem  = posA % 196;              // y*14 + x
    const int ncol = nbase + mrow;            // B-col = output index

    v8f acc = {};
    #pragma unroll
    for (int kb = 0; kb < 8; ++kb) {
        const int k0 = kb * 4 + 2 * hi2;
        v2f av, bv;
        av.x = h[((size_t)bA * 32 + k0    ) * 196 + rem];
        av.y = h[((size_t)bA * 32 + k0 + 1) * 196 + rem];
        bv.x = Wsrc[(size_t)ncol * 32 + k0    ];
        bv.y = Wsrc[(size_t)ncol * 32 + k0 + 1];
        acc = __builtin_amdgcn_wmma_f32_16x16x4_f32(false, av, false, bv,
                                                    (short)0, acc, false, false);
    }

    #pragma unroll
    for (int r = 0; r < 8; ++r) {
        const int M   = r + 8 * hi2;
        const int N   = lane & 15;
        const int pos = tile * 16 + M;
        const int o   = nbase + N;
        const float v = acc[r];
        if (isAct) {
            a1[(size_t)pos * 32 + o] = 1.0f / (1.0f + expf(-(v + act_b[o])));
        } else {
            const int cap = o >> 4, pp = o & 15;
            M1[((size_t)pos * 32 + cap) * 16 + pp] = v + pose_b[o];
        }
    }
}

// ---------------------------------------------------------------------------
// V[i][j][p] generator: 4x4 pose matmul, weights streamed from L2.
// MODE 0 (conv caps): i = (x*3+y)*32+ci, M from [b][hi][wi][ci][16],
//                     W row = Wv[(i*J + j)*16 ...]
// MODE 1 (fc caps)  : i = (hh*4+ww)*32+ci, M from [b*IC + i][16],
//                     W row = Wv[((i&31)*J + j)*16 ...], + receptive centers
// ---------------------------------------------------------------------------
template<int IC, int J, int MODE>
__device__ __forceinline__ void get_v(const float* __restrict__ Mg,
                                      const float* __restrict__ Wv,
                                      int b, int hp, int wp,
                                      int Hin, int Win, int stride,
                                      int i, int j, float* V)
{
    const float* m;
    const float* w;
    if (MODE == 0) {
        int grp = i >> 5, ci = i & 31;
        int xx = grp / 3, yy = grp - xx * 3;
        int hi = hp * stride + xx, wi = wp * stride + yy;
        m = Mg + ((((size_t)b * Hin + hi) * Win + wi) * 32 + ci) * 16;
        w = Wv + ((size_t)i * J + j) * 16;
    } else {
        m = Mg + ((size_t)b * IC + i) * 16;
        w = Wv + ((size_t)(i & 31) * J + j) * 16;
    }
    #pragma unroll
    for (int a = 0; a < 4; ++a) {
        float m0 = m[a*4+0], m1 = m[a*4+1], m2 = m[a*4+2], m3 = m[a*4+3];
        #pragma unroll
        for (int c = 0; c < 4; ++c) {
            V[a*4+c] = fmaf(m0, w[c], fmaf(m1, w[4+c], fmaf(m2, w[8+c], m3 * w[12+c])));
        }
    }
    if (MODE == 1) {
        int hh = i >> 7, ww = (i >> 5) & 3;
        V[3] += (8.5f + 4.0f * (float)hh) * 0.03125f;   // xs[h]/32 at pose (0,3)
        V[7] += (8.5f + 4.0f * (float)ww) * 0.03125f;   // ys[w]/32 at pose (1,3)
    }
}

// ---------------------------------------------------------------------------
// Fused EM-routing kernel: one workgroup per output position.
// All routing state (R, mu, sigma2, a) LDS-resident; V recomputed per pass.
// Deterministic fixed-order reductions via per-group partials.
// ---------------------------------------------------------------------------
template<int IC, int J, int NG, int MODE>
__global__ __launch_bounds__(J * NG)
void em_caps_kernel(const float* __restrict__ Mg, const float* __restrict__ ag,
                    const float* __restrict__ Wv,
                    const float* __restrict__ beta_u, const float* __restrict__ beta_a,
                    float* __restrict__ Mout, float* __restrict__ aout,
                    int Hin, int Win, int Hp, int Wp, int stride)
{
    constexpr int BLK = J * NG;
    __shared__ float sA[IC];
    __shared__ float sR[IC * J];
    __shared__ float sPart[NG * J * 16];
    __shared__ float sPartS[NG * J];
    __shared__ float sMu[J * 16];
    __shared__ float sSig[J * 16];
    __shared__ float sSR[J];
    __shared__ float sAout[J];
    __shared__ float sLogA[J];
    __shared__ float sLogSig[J];

    const int tid = threadIdx.x;
    const int wg  = blockIdx.x;
    const int wp  = wg % Wp;
    const int hp  = (wg / Wp) % Hp;
    const int b   = wg / (Wp * Hp);

    // Load input activations; init R = 1/J.
    for (int i = tid; i < IC; i += BLK) {
        if (MODE == 0) {
            int grp = i >> 5, ci = i & 31;
            int xx = grp / 3, yy = grp - xx * 3;
            int hi = hp * stride + xx, wi = wp * stride + yy;
            sA[i] = ag[(((size_t)b * Hin + hi) * Win + wi) * 32 + ci];
        } else {
            sA[i] = ag[(size_t)b * IC + i];
        }
    }
    for (int idx = tid; idx < IC * J; idx += BLK) sR[idx] = 1.0f / (float)J;
    __syncthreads();

    const int j = tid % J;
    const int g = tid / J;

    float q = 1.0f;
    for (int it = 0; it < 3; ++it) {
        q *= 0.95f;
        const float lam = 0.01f * (1.0f - q);

        // ---- Pass A: sR[j] = sum_i Ra + eps ; mu numerators ----
        {
            float accP[16]; float accS = 0.0f;
            #pragma unroll
            for (int p = 0; p < 16; ++p) accP[p] = 0.0f;
            for (int i = g; i < IC; i += NG) {
                float V[16];
                get_v<IC, J, MODE>(Mg, Wv, b, hp, wp, Hin, Win, stride, i, j, V);
                float ra = sR[i * J + j] * sA[i];
                accS += ra;
                #pragma unroll
                for (int p = 0; p < 16; ++p) accP[p] = fmaf(ra, V[p], accP[p]);
            }
            sPartS[g * J + j] = accS;
            #pragma unroll
            for (int p = 0; p < 16; ++p) sPart[(g * J + j) * 16 + p] = accP[p];
        }
        __syncthreads();
        if (tid < J) {
            float s = EPSV;
            #pragma unroll
            for (int g2 = 0; g2 < NG; ++g2) s += sPartS[g2 * J + tid];
            sSR[tid] = s;
        }
        __syncthreads();
        for (int idx = tid; idx < J * 16; idx += BLK) {
            int jj = idx >> 4;
            float s = 0.0f;
            #pragma unroll
            for (int g2 = 0; g2 < NG; ++g2) s += sPart[(g2 * J + jj) * 16 + (idx & 15)];
            sMu[idx] = s / sSR[jj];
        }
        __syncthreads();

        // ---- Pass B: sigma2 numerators ----
        {
            float muj[16];
            #pragma unroll
            for (int p = 0; p < 16; ++p) muj[p] = sMu[j * 16 + p];
            float accP[16];
            #pragma unroll
            for (int p = 0; p < 16; ++p) accP[p] = 0.0f;
            for (int i = g; i < IC; i += NG) {
                float V[16];
                get_v<IC, J, MODE>(Mg, Wv, b, hp, wp, Hin, Win, stride, i, j, V);
                float ra = sR[i * J + j] * sA[i];
                #pragma unroll
                for (int p = 0; p < 16; ++p) {
                    float d = V[p] - muj[p];
                    accP[p] = fmaf(ra, d * d, accP[p]);
                }
            }
            #pragma unroll
            for (int p = 0; p < 16; ++p) sPart[(g * J + j) * 16 + p] = accP[p];
        }
        __syncthreads();
        for (int idx = tid; idx < J * 16; idx += BLK) {
            int jj = idx >> 4;
            float s = 0.0f;
            #pragma unroll
            for (int g2 = 0; g2 < NG; ++g2) s += sPart[(g2 * J + jj) * 16 + (idx & 15)];
            sSig[idx] = s / sSR[jj] + EPSV;
        }
        __syncthreads();
        if (tid < J) {
            float ls = 0.0f;
            #pragma unroll
            for (int p = 0; p < 16; ++p) ls += logf(sSig[tid * 16 + p]);
            float cost = (16.0f * beta_u[tid] + 0.5f * ls) * sSR[tid];
            float aa = 1.0f / (1.0f + expf(-lam * (beta_a[tid] - cost)));
            sAout[tid]   = aa;
            sLogA[tid]   = logf(aa + EPSV);
            sLogSig[tid] = ls;
        }
        __syncthreads();

        // ---- Pass C: new responsibilities (skip on last iteration) ----
        if (it < 2) {
            float muj[16], sgj[16];
            #pragma unroll
            for (int p = 0; p < 16; ++p) { muj[p] = sMu[j*16+p]; sgj[p] = sSig[j*16+p]; }
            const float halfls = 0.5f * sLogSig[j];
            const float la     = sLogA[j];
            for (int i = g; i < IC; i += NG) {
                float V[16];
                get_v<IC, J, MODE>(Mg, Wv, b, hp, wp, Hin, Win, stride, i, j, V);
                float dsum = 0.0f;
                #pragma unroll
                for (int p = 0; p < 16; ++p) {
                    float d = V[p] - muj[p];
                    dsum += d * d / sgj[p];
                }
                sR[i * J + j] = -0.5f * dsum - halfls + la;
            }
            __syncthreads();
            // softmax over j, per input capsule i
            for (int i = tid; i < IC; i += BLK) {
                float lv[J];
                float mx = -3.4e38f;
                #pragma unroll
                for (int j2 = 0; j2 < J; ++j2) { lv[j2] = sR[i*J+j2]; mx = fmaxf(mx, lv[j2]); }
                float s = 0.0f;
                #pragma unroll
                for (int j2 = 0; j2 < J; ++j2) { lv[j2] = expf(lv[j2] - mx); s += lv[j2]; }
                float inv = 1.0f / s;
                #pragma unroll
                for (int j2 = 0; j2 < J; ++j2) sR[i*J+j2] = lv[j2] * inv;
            }
            __syncthreads();
        }
    }

    // Write outputs: Mout[pos][J][16], aout[pos][J]
    const size_t pos = (size_t)(b * Hp + hp) * Wp + wp;
    for (int idx = tid; idx < J * 16; idx += BLK) Mout[pos * J * 16 + idx] = sMu[idx];
    if (tid < J) aout[pos * J + tid] = sAout[tid];
}

// ---------------------------------------------------------------------------
extern "C" void kernel_launch(void* const* d_in, const int* in_sizes, int n_in,
                              void* d_out, int out_size, void* d_ws, size_t ws_size,
                              hipStream_t stream)
{
    (void)in_sizes; (void)n_in; (void)out_size; (void)ws_size;

    const float* x         = (const float*)d_in[0];
    const float* conv1_w   = (const float*)d_in[1];
    const float* conv1_b   = (const float*)d_in[2];
    const float* pc_act_w  = (const float*)d_in[3];
    const float* pc_act_b  = (const float*)d_in[4];
    const float* pc_pose_w = (const float*)d_in[5];
    const float* pc_pose_b = (const float*)d_in[6];
    const float* W1        = (const float*)d_in[7];
    const float* beta_u1   = (const float*)d_in[8];
    const float* beta_a1   = (const float*)d_in[9];
    const float* W2        = (const float*)d_in[10];
    const float* beta_u2   = (const float*)d_in[11];
    const float* beta_a2   = (const float*)d_in[12];
    const float* Wfc       = (const float*)d_in[13];
    const float* beta_ufc  = (const float*)d_in[14];
    const float* beta_afc  = (const float*)d_in[15];

    float* out = (float*)d_out;
    float* ws  = (float*)d_ws;

    float* h  = ws;                 // 8*32*14*14          = 50176
    float* M1 = h  + 50176;         // 8*14*14*32*16       = 802816
    float* a1 = M1 + 802816;        // 8*14*14*32          = 50176
    float* M2 = a1 + 50176;         // 8*6*6*32*16         = 147456
    float* a2 = M2 + 147456;        // 8*6*6*32            = 9216
    float* M3 = a2 + 9216;          // 8*4*4*32*16         = 65536
    float* a3 = M3 + 65536;         // 8*4*4*32            = 4096

    // 1. conv1 + relu
    conv1_relu_kernel<<<196, 256, 0, stream>>>(x, conv1_w, conv1_b, h);

    // 2. primary caps (WMMA f32 16x16x4): pose tiles y=0..31, act tiles y=32..33
    primary_caps_kernel<<<dim3(98, 34), 32, 0, stream>>>(
        h, pc_pose_w, pc_pose_b, pc_act_w, pc_act_b, M1, a1);

    // 3. conv caps stage 1: 14x14 -> 6x6, k=3, s=2, I=288, J=32
    em_caps_kernel<288, 32, 8, 0><<<8 * 6 * 6, 256, 0, stream>>>(
        M1, a1, W1, beta_u1, beta_a1, M2, a2, 14, 14, 6, 6, 2);

    // 4. conv caps stage 2: 6x6 -> 4x4, k=3, s=1, I=288, J=32
    em_caps_kernel<288, 32, 8, 0><<<8 * 4 * 4, 256, 0, stream>>>(
        M2, a2, W2, beta_u2, beta_a2, M3, a3, 6, 6, 4, 4, 1);

    // 5. FC caps: I=512, J=10, + receptive centers; writes mu then a into d_out
    em_caps_kernel<512, 10, 16, 1><<<8, 160, 0, stream>>>(
        M3, a3, Wfc, beta_ufc, beta_afc, out, out + 1280, 4, 4, 1, 1, 1);
}